// LinAminoToAtom_28028956574045
// MI455X (gfx1250) — compile-verified
//
#include <hip/hip_runtime.h>
#include <hip/hip_bf16.h>

// ---- static problem config (mirrors the reference) ----
#define BATCH 64
#define NODES 512
#define AMINO 64
#define ATOM  32
#define RMAX  13
#define JCOLS (RMAX * ATOM)   // 416 output columns per node
#define MROWS (BATCH * 3)     // 192 rows: m = v*64 + b  (xyz-major)
#define MPAD  (MROWS + 8)     // LDS row padding
#define MT    (MROWS / 16)    // 12 M tiles (4 tiles per xyz component)
#define MHALF (MT / 2)        // 6 M tiles per work item
#define NPAIR (JCOLS / 32)    // 13 residue slots (N-tile pairs)
#define KSTEPS (AMINO / 4)    // 16 WMMA K-steps

typedef float v2f __attribute__((ext_vector_type(2)));
typedef float v8f __attribute__((ext_vector_type(8)));

// ---- tiny helpers: build inverse gather map in workspace ----
__global__ void lin_init_inv(int* __restrict__ inv, int n) {
    int i = blockIdx.x * blockDim.x + threadIdx.x;
    if (i < n) inv[i] = -1;
}

__global__ void lin_scatter_inv(const int* __restrict__ gather_idx,
                                int* __restrict__ inv, int total) {
    int p = blockIdx.x * blockDim.x + threadIdx.x;
    if (p < total) inv[gather_idx[p]] = p;
}

// ---- main kernel: one node per workgroup, 8 waves, B-resident f32 WMMA ----
__global__ __launch_bounds__(256)
void lin_amino_to_atom_wmma(const float* __restrict__ x,
                            const float* __restrict__ W,
                            const int*   __restrict__ type_ids,
                            const int*   __restrict__ inv,
                            float*       __restrict__ out,
                            int total_atoms) {
    // xs[a][m], m = v*64 + b: A-matrix, K(amino)-major; the xyz-major row
    // mapping makes every 16-row M tile single-v -> division-free stores.
    __shared__ float xs[AMINO][MPAD];   // 64 * 200 * 4B = 51.2 KB of 320 KB LDS

    const int n   = blockIdx.x;
    const int tid = threadIdx.x;
    const int t   = type_ids[n];

    const float* __restrict__ Wt = W + (size_t)t * AMINO * JCOLS;

    // Warm L2/L0 for this node's weight block (64*416*4B = 106.5 KB).
    __builtin_prefetch(Wt + (size_t)tid * 104, 0, 1);

    // Stage x[:, n, :, :] into LDS: xs[a][v*64 + b] = x[b][n][a][v].
    // One (b,a) pair per thread: consecutive lanes read consecutive 12B
    // triples (coalesced); LDS writes are 3 stride-64 dwords, conflict-free.
    for (int idx = tid; idx < BATCH * AMINO; idx += 256) {
        const int b = idx >> 6;              // AMINO == 64
        const int a = idx & 63;
        const float* __restrict__ src =
            x + ((size_t)((size_t)b * NODES + n) * AMINO + a) * 3;
        const float v0 = src[0], v1 = src[1], v2 = src[2];
        xs[a][b]       = v0;
        xs[a][b + 64]  = v1;
        xs[a][b + 128] = v2;
    }
    __syncthreads();

    const int lane = tid & 31;
    const int wave = tid >> 5;
    const int lgrp = lane >> 4;     // 0: lanes 0-15, 1: lanes 16-31
    const int lidx = lane & 15;

    const size_t bstride = (size_t)total_atoms * (ATOM * 3); // floats per b

    // Work item = (residue slot, M half). B fragments for the slot's 32
    // atom columns are loaded ONCE into registers (64 VGPRs) and reused by
    // 6 M tiles, so the steady-state M loop touches only LDS + XDL.
    // All loop/branch conditions are wave-uniform -> EXEC all-ones at WMMA.
    for (int item = wave; item < NPAIR * 2; item += 8) {
        const int s  = item >> 1;            // residue slot jhi
        const int mh = item & 1;             // which half of the 12 M tiles

        // Skip residues dropped by the gather: ~40% of slots never compute.
        const int p = inv[n * RMAX + s];     // wave-uniform
        if (p < 0) continue;

        const int nb0 = s * 32;              // atoms 0..15 of this residue
        const int nb1 = nb0 + 16;            // atoms 16..31

        // B fragments (4x16 each K-step): lanes 0-15 hold K=ka,ka+1 rows,
        // lanes 16-31 hold K=ka+2,ka+3 rows, N striped across lanes.
        v2f bf0[KSTEPS], bf1[KSTEPS];
        #pragma unroll
        for (int ks = 0; ks < KSTEPS; ++ks) {
            const int ka = 4 * ks + 2 * lgrp;
            const float* __restrict__ wr = Wt + (size_t)ka * JCOLS;
            bf0[ks].x = wr[nb0 + lidx];
            bf0[ks].y = wr[JCOLS + nb0 + lidx];
            bf1[ks].x = wr[nb1 + lidx];
            bf1[ks].y = wr[JCOLS + nb1 + lidx];
        }

        for (int mt = mh * MHALF; mt < (mh + 1) * MHALF; ++mt) {
            const int mb = mt * 16;
            v8f acc0 = {}, acc1 = {};
            #pragma unroll
            for (int ks = 0; ks < KSTEPS; ++ks) {
                // A fragment (16x4): lanes 0-15 K=ka,ka+1; 16-31 K=ka+2,ka+3
                const int ka = 4 * ks + 2 * lgrp;
                v2f af;
                af.x = xs[ka    ][mb + lidx];
                af.y = xs[ka + 1][mb + lidx];

                // Two independent accumulation chains share the A fragment.
                acc0 = __builtin_amdgcn_wmma_f32_16x16x4_f32(
                    false, af, false, bf0[ks], (short)0, acc0, false, false);
                acc1 = __builtin_amdgcn_wmma_f32_16x16x4_f32(
                    false, af, false, bf1[ks], (short)0, acc1, false, false);
            }

            // Division-free scatter: whole tile is one xyz component.
            // out[((b*total_atoms + p)*32 + atom)*3 + v], b = b0 + r.
            const int vcoord = mb >> 6;                 // xyz component
            const int b0     = (mb & 63) + 8 * lgrp;    // C/D: M=r / M=r+8
            float* __restrict__ o =
                out + ((size_t)b0 * total_atoms + p) * (ATOM * 3)
                    + lidx * 3 + vcoord;
            #pragma unroll
            for (int r = 0; r < 8; ++r) {
                o[(size_t)r * bstride]      = acc0[r];  // atoms  0..15
                o[(size_t)r * bstride + 48] = acc1[r];  // atoms 16..31
            }
        }
    }
}

extern "C" void kernel_launch(void* const* d_in, const int* in_sizes, int n_in,
                              void* d_out, int out_size, void* d_ws, size_t ws_size,
                              hipStream_t stream) {
    const float* x          = (const float*)d_in[0];
    const float* W          = (const float*)d_in[1];
    const int*   type_ids   = (const int*)d_in[2];
    const int*   gather_idx = (const int*)d_in[3];
    const int total_atoms   = in_sizes[3];
    float* out = (float*)d_out;

    int* inv = (int*)d_ws;                 // NODES*RMAX = 6656 ints
    const int npad = NODES * RMAX;

    lin_init_inv<<<(npad + 255) / 256, 256, 0, stream>>>(inv, npad);
    lin_scatter_inv<<<(total_atoms + 255) / 256, 256, 0, stream>>>(gather_idx, inv,
                                                                   total_atoms);
    lin_amino_to_atom_wmma<<<NODES, 256, 0, stream>>>(x, W, type_ids, inv, out,
                                                      total_atoms);
}